// TwoStageGNN_22651657519255
// MI455X (gfx1250) — compile-verified
//
#include <hip/hip_runtime.h>
#include <hip/hip_bf16.h>
#include <math.h>

#define N_ATOMS   20000
#define E_ATOM    160000
#define N_POLY    2000
#define E_POLY    16000
#define N_CRYSTAL 16
#define D_IN      92
#define Hd        128
#define HEADS     4
#define EDGE_DIM  4
#define N_LAYERS  3
#define ET        (E_ATOM + N_ATOMS)   // edges + self loops
#define KC        32                   // GEMM K-chunk staged in LDS

typedef __attribute__((ext_vector_type(2))) float v2f;
typedef __attribute__((ext_vector_type(8))) float v8f;
typedef unsigned int u32x4 __attribute__((ext_vector_type(4)));
typedef int i32x4 __attribute__((ext_vector_type(4)));
typedef int i32x8 __attribute__((ext_vector_type(8)));

#if __has_builtin(__builtin_amdgcn_tensor_load_to_lds) && __has_builtin(__builtin_amdgcn_s_wait_tensorcnt)
#define USE_TDM 1
#else
#define USE_TDM 0
#endif

static __device__ __forceinline__ float atomicMaxF(float* addr, float value) {
    if (value >= 0.0f)
        return __int_as_float(atomicMax((int*)addr, __float_as_int(value)));
    else
        return __uint_as_float(atomicMin((unsigned int*)addr, __float_as_uint(value)));
}

__global__ void k_fill(float* __restrict__ p, float v, int n) {
    int i = blockIdx.x * blockDim.x + threadIdx.x;
    if (i < n) p[i] = v;
}

#if USE_TDM
// LDS byte offset of a __shared__ object (generic -> AS(3) -> int)
static __device__ __forceinline__ unsigned lds_offset(const void* p) {
    return (unsigned)(__UINTPTR_TYPE__)(__attribute__((address_space(3))) const void*)p;
}

// Issue a TDM 2-D tile load: W[kb:kb+kc, n0:n0+32] (row-major, f32) -> LDS.
// D# packing per CDNA5 ISA 8.3/8.4 (group0: count/lds/global/type,
// group1: data_size, tensor_dim0/1, tile_dim0/1, dim0 stride).
static __device__ __forceinline__ void tdm_load_b_panel(const float* W, int K, int N,
                                                        int n0, int kb, int kc,
                                                        unsigned lds) {
    unsigned long long ga =
        (unsigned long long)(__UINTPTR_TYPE__)(W + (size_t)kb * N + n0);
    u32x4 g0;
    g0[0] = 1u;                                               // count=1, no gather
    g0[1] = lds;                                              // lds_addr (bytes)
    g0[2] = (unsigned)ga;                                     // global_addr[31:0]
    g0[3] = (unsigned)((ga >> 32) & 0x01FFFFFFu) | (2u << 30); // addr[56:32] | type=2
    i32x8 g1;
    g1[0] = (int)(2u << 16);                                  // data_size = 4B
    g1[1] = (int)(((unsigned)N & 0xFFFFu) << 16);             // tensor_dim0 lo16 @48
    g1[2] = (int)(((unsigned)N >> 16) | (((unsigned)K & 0xFFFFu) << 16)); // d0 hi | d1 lo
    g1[3] = (int)(((unsigned)K >> 16) | (32u << 16));         // d1 hi | tile_dim0=32
    g1[4] = kc;                                               // tile_dim1
    g1[5] = N;                                                // tensor_dim0_stride lo32
    g1[6] = 0;
    g1[7] = 0;
    i32x4 z4 = {0, 0, 0, 0};                                  // 2-D: groups 2/3 unused
    i32x8 z8 = {0, 0, 0, 0, 0, 0, 0, 0};
    // 6-arg form (clang-23 / therock-10.0 headers)
    __builtin_amdgcn_tensor_load_to_lds(g0, g1, z4, z4, z8, 0);
}
#endif

// ---------------------------------------------------------------------------
// f32 WMMA GEMM:  C[M,N] = act(A[M,K] @ W[K,N] + bias), row-major.
// Requires M%16==0, N%32==0, K%4==0. Block = 64 threads (2 waves), block tile
// 16x32; each wave one 16x16 WMMA tile. B panels staged in LDS by the Tensor
// Data Mover (double-buffered, s_wait_tensorcnt), A via coalesced b64 loads.
// Native f32 WMMA: model is f32 end-to-end and scatter-bound, so no
// down-conversion is warranted.
// ---------------------------------------------------------------------------
__global__ void k_wmma_gemm(const float* __restrict__ A, const float* __restrict__ W,
                            const float* __restrict__ bias, float* __restrict__ C,
                            int M, int K, int N, int relu) {
    __shared__ float bufB[2][KC * 32];
    const int tid  = threadIdx.x;
    const int wv   = tid >> 5;        // wave in block: 0..1
    const int lane = tid & 31;
    const int half = lane >> 4;       // 0: K {0,1}/rows m..m+7 ; 1: K {2,3}/m+8..
    const int l16  = lane & 15;
    const int n0b  = blockIdx.x * 32; // block col base
    const int m0   = blockIdx.y * 16;

    v8f acc = {};
    const float* arow = A + (size_t)(m0 + l16) * K + 2 * half;

#if USE_TDM
    if (wv == 0) {
        int kc0 = K < KC ? K : KC;
        tdm_load_b_panel(W, K, N, n0b, 0, kc0, lds_offset(&bufB[0][0]));
    }
#endif
    int cur = 0;
    for (int kb = 0; kb < K; kb += KC, cur ^= 1) {
        const int kc = (K - kb) < KC ? (K - kb) : KC;
#if USE_TDM
        if (wv == 0) {
            int kn = kb + KC;
            if (kn < K) {
                int kcn = (K - kn) < KC ? (K - kn) : KC;
                tdm_load_b_panel(W, K, N, n0b, kn, kcn, lds_offset(&bufB[cur ^ 1][0]));
                __builtin_amdgcn_s_wait_tensorcnt((short)1);  // chunk kb complete
            } else {
                __builtin_amdgcn_s_wait_tensorcnt((short)0);
            }
        }
        __syncthreads();              // panel [cur] visible to both waves
#else
        __syncthreads();              // protect buffer reuse
        for (int idx = tid; idx < kc * 32; idx += 64) {
            int r = idx >> 5, c = idx & 31;
            bufB[cur][idx] = W[(size_t)(kb + r) * N + n0b + c];
        }
        __syncthreads();
#endif
        const float* abase = arow + kb;
        const float* bb0   = &bufB[cur][2 * half * 32 + wv * 16 + l16];
        if (kc == KC) {
            __builtin_prefetch(abase + KC, 0, 0);             // next A chunk
#pragma unroll
            for (int k = 0; k < KC; k += 4) {
                v2f a, b;
                a.x = abase[k];
                a.y = abase[k + 1];
                b.x = bb0[(k + 0) * 32];                      // ds_load from LDS panel
                b.y = bb0[(k + 1) * 32];
                acc = __builtin_amdgcn_wmma_f32_16x16x4_f32(false, a, false, b,
                                                            (short)0, acc, false, false);
            }
        } else {
            for (int k = 0; k < kc; k += 4) {
                v2f a, b;
                a.x = abase[k];
                a.y = abase[k + 1];
                b.x = bb0[(k + 0) * 32];
                b.y = bb0[(k + 1) * 32];
                acc = __builtin_amdgcn_wmma_f32_16x16x4_f32(false, a, false, b,
                                                            (short)0, acc, false, false);
            }
        }
        __syncthreads();              // both waves done with [cur] before TDM reuses it
    }

    const int nc = n0b + wv * 16;
    const float bb = bias ? bias[nc + l16] : 0.0f;
#pragma unroll
    for (int v = 0; v < 8; ++v) {     // C: VGPR v -> row m0 + 8*half + v
        float r = acc[v] + bb;
        if (relu) r = fmaxf(r, 0.0f);
        C[(size_t)(m0 + 8 * half + v) * N + nc + l16] = r;
    }
}

// GEMV for N==1 layers: out[i] = A[i,:] . w + b
__global__ void k_gemv(const float* __restrict__ A, const float* __restrict__ w,
                       const float* __restrict__ b, float* __restrict__ out,
                       int M, int K) {
    int i = blockIdx.x * blockDim.x + threadIdx.x;
    if (i >= M) return;
    const float* r = A + (size_t)i * K;
    float s = b ? b[0] : 0.0f;
    for (int k = 0; k < K; ++k) s += r[k] * w[k];
    out[i] = s;
}

// ------------------------------ GAT stage ----------------------------------
__global__ void k_gat_scores(const float* __restrict__ wx, const float* __restrict__ a_src,
                             const float* __restrict__ a_dst, float* __restrict__ as_,
                             float* __restrict__ ad_) {
    int i = blockIdx.x * blockDim.x + threadIdx.x;     // n*HEADS + h
    if (i >= N_ATOMS * HEADS) return;
    int h = i & (HEADS - 1), n = i >> 2;
    const float* row = wx + (size_t)n * (HEADS * Hd) + (size_t)h * Hd;
    const float* vs = a_src + h * Hd;
    const float* vd = a_dst + h * Hd;
    float s = 0.f, d = 0.f;
    for (int j = 0; j < Hd; ++j) { float x = row[j]; s += x * vs[j]; d += x * vd[j]; }
    as_[i] = s; ad_[i] = d;
}

static __device__ __forceinline__ void edge_sd(const int* __restrict__ ei, int e,
                                               int& s, int& d) {
    if (e < E_ATOM) { s = ei[e]; d = ei[E_ATOM + e]; }
    else            { s = d = e - E_ATOM; }            // self loop
}

static __device__ __forceinline__ float leaky02(float x) {
    return x > 0.0f ? x : 0.2f * x;
}

__global__ void k_edge_max(const int* __restrict__ ei, const float* __restrict__ as_,
                           const float* __restrict__ ad_, float* __restrict__ segmax) {
    int i = blockIdx.x * blockDim.x + threadIdx.x;     // e*HEADS + h
    if (i >= ET * HEADS) return;
    int h = i & (HEADS - 1), e = i >> 2, s, d;
    edge_sd(ei, e, s, d);
    float sc = leaky02(as_[s * HEADS + h] + ad_[d * HEADS + h]);
    atomicMaxF(&segmax[d * HEADS + h], sc);
}

__global__ void k_edge_exp(const int* __restrict__ ei, const float* __restrict__ as_,
                           const float* __restrict__ ad_, const float* __restrict__ segmax,
                           float* __restrict__ segsum, float* __restrict__ ealpha) {
    int i = blockIdx.x * blockDim.x + threadIdx.x;
    if (i >= ET * HEADS) return;
    int h = i & (HEADS - 1), e = i >> 2, s, d;
    edge_sd(ei, e, s, d);
    float sc = leaky02(as_[s * HEADS + h] + ad_[d * HEADS + h]);
    float ex = expf(sc - segmax[d * HEADS + h]);
    ealpha[i] = ex;
    atomicAdd(&segsum[d * HEADS + h], ex);
}

// head-mean folded in (x0.25), one atomicAdd per (edge, feature)
__global__ void k_edge_aggr(const int* __restrict__ ei, const float* __restrict__ ealpha,
                            const float* __restrict__ segsum, const float* __restrict__ wx,
                            float* __restrict__ hn) {
    long long i = (long long)blockIdx.x * blockDim.x + threadIdx.x; // e*Hd + d
    if (i >= (long long)ET * Hd) return;
    int dft = (int)(i & (Hd - 1)), e = (int)(i >> 7), s, d;
    edge_sd(ei, e, s, d);
    float acc = 0.f;
#pragma unroll
    for (int h = 0; h < HEADS; ++h) {
        float a = ealpha[e * HEADS + h] / (segsum[d * HEADS + h] + 1e-16f);
        acc += a * wx[(size_t)s * (HEADS * Hd) + h * Hd + dft];
    }
    atomicAdd(&hn[(size_t)d * Hd + dft], 0.25f * acc);
}

// h = relu(LayerNorm(x + pb)) + h ; one wave32 per 128-wide row
__global__ void k_ln_relu_res(const float* __restrict__ xin, const float* __restrict__ pb,
                              const float* __restrict__ gg, const float* __restrict__ be,
                              float* __restrict__ hio, int M) {
    int row  = (int)((blockIdx.x * blockDim.x + threadIdx.x) >> 5);
    int lane = threadIdx.x & 31;
    if (row >= M) return;
    const float* x = xin + (size_t)row * Hd;
    float v[4], s = 0.f;
#pragma unroll
    for (int j = 0; j < 4; ++j) {
        int d = lane + 32 * j;
        v[j] = x[d] + (pb ? pb[d] : 0.f);
        s += v[j];
    }
#pragma unroll
    for (int o = 16; o > 0; o >>= 1) s += __shfl_xor(s, o, 32);
    float mean = s * (1.0f / Hd);
    float q = 0.f;
#pragma unroll
    for (int j = 0; j < 4; ++j) { float t = v[j] - mean; q += t * t; }
#pragma unroll
    for (int o = 16; o > 0; o >>= 1) q += __shfl_xor(q, o, 32);
    float rstd = rsqrtf(q * (1.0f / Hd) + 1e-5f);
    float* out = hio + (size_t)row * Hd;
#pragma unroll
    for (int j = 0; j < 4; ++j) {
        int d = lane + 32 * j;
        float y = (v[j] - mean) * rstd * gg[d] + be[d];
        out[d] = fmaxf(y, 0.f) + out[d];
    }
}

// ------------------------- attention pooling (stage 1) ----------------------
__global__ void k_reduce_max(const float* __restrict__ s, float* __restrict__ red, int n) {
    float m = -__builtin_inff();
    for (int i = blockIdx.x * blockDim.x + threadIdx.x; i < n; i += gridDim.x * blockDim.x)
        m = fmaxf(m, s[i]);
#pragma unroll
    for (int o = 16; o > 0; o >>= 1) m = fmaxf(m, __shfl_xor(m, o, 32));
    if ((threadIdx.x & 31) == 0) atomicMaxF(&red[0], m);
}

__global__ void k_exp_sum(float* __restrict__ s, float* __restrict__ red, int n) {
    int i = blockIdx.x * blockDim.x + threadIdx.x;
    if (i >= n) return;
    float e = expf(s[i] - red[0]);
    s[i] = e;
    atomicAdd(&red[1], e);
}

__global__ void k_attpool(const float* __restrict__ h, const float* __restrict__ attw,
                          const float* __restrict__ red, const int* __restrict__ pbi,
                          float* __restrict__ poly) {
    long long i = (long long)blockIdx.x * blockDim.x + threadIdx.x; // n*Hd + d
    if (i >= (long long)N_ATOMS * Hd) return;
    int d = (int)(i & (Hd - 1)), n = (int)(i >> 7);
    atomicAdd(&poly[(size_t)pbi[n] * Hd + d], h[i] * attw[n] / red[1]);
}

// ------------------------------ ECC stage ----------------------------------
__global__ void k_cat260(const float* __restrict__ g, const int* __restrict__ iei,
                         const float* __restrict__ eattr, float* __restrict__ cat) {
    long long i = (long long)blockIdx.x * blockDim.x + threadIdx.x;
    if (i >= (long long)E_POLY * (2 * Hd + EDGE_DIM)) return;
    int c = (int)(i % (2 * Hd + EDGE_DIM));
    int e = (int)(i / (2 * Hd + EDGE_DIM));
    int s = iei[e], d = iei[E_POLY + e];
    float v;
    if (c < Hd)            v = g[(size_t)d * Hd + c];            // x_i (target)
    else if (c < 2 * Hd)   v = g[(size_t)s * Hd + (c - Hd)];     // x_j (source)
    else                   v = eattr[(size_t)e * EDGE_DIM + (c - 2 * Hd)];
    cat[i] = v;
}

__global__ void k_scatter_add(const int* __restrict__ iei, const float* __restrict__ msg,
                              float* __restrict__ aggr) {
    long long i = (long long)blockIdx.x * blockDim.x + threadIdx.x; // e*Hd + d
    if (i >= (long long)E_POLY * Hd) return;
    int d = (int)(i & (Hd - 1)), e = (int)(i >> 7);
    atomicAdd(&aggr[(size_t)iei[E_POLY + e] * Hd + d], msg[i]);
}

__global__ void k_cat256(const float* __restrict__ g, const float* __restrict__ aggr,
                         float* __restrict__ cat) {
    int i = blockIdx.x * blockDim.x + threadIdx.x;
    if (i >= N_POLY * 2 * Hd) return;
    int c = i & (2 * Hd - 1), p = i >> 8;
    cat[i] = (c < Hd) ? g[(size_t)p * Hd + c] : aggr[(size_t)p * Hd + (c - Hd)];
}

// --------------------------- crystal pooling -------------------------------
__global__ void k_seg_max(const float* __restrict__ s, const int* __restrict__ cb,
                          float* __restrict__ cmax, int n) {
    int i = blockIdx.x * blockDim.x + threadIdx.x;
    if (i < n) atomicMaxF(&cmax[cb[i]], s[i]);
}
__global__ void k_fix_max(float* __restrict__ cmax, int n) {
    int i = threadIdx.x;
    if (i < n && !isfinite(cmax[i])) cmax[i] = 0.0f;
}
__global__ void k_seg_exp(float* __restrict__ s, const int* __restrict__ cb,
                          const float* __restrict__ cmax, float* __restrict__ csum, int n) {
    int i = blockIdx.x * blockDim.x + threadIdx.x;
    if (i >= n) return;
    float e = expf(s[i] - cmax[cb[i]]);
    s[i] = e;
    atomicAdd(&csum[cb[i]], e);
}
__global__ void k_cpool(const float* __restrict__ g, const float* __restrict__ ew,
                        const float* __restrict__ csum, const int* __restrict__ cb,
                        float* __restrict__ graph) {
    int i = blockIdx.x * blockDim.x + threadIdx.x;   // p*Hd + d
    if (i >= N_POLY * Hd) return;
    int d = i & (Hd - 1), p = i >> 7;
    atomicAdd(&graph[(size_t)cb[p] * Hd + d], g[i] * ew[p] / (csum[cb[p]] + 1e-16f));
}

// ===========================================================================
extern "C" void kernel_launch(void* const* d_in, const int* in_sizes, int n_in,
                              void* d_out, int out_size, void* d_ws, size_t ws_size,
                              hipStream_t stream) {
    (void)in_sizes; (void)n_in; (void)out_size; (void)ws_size;
    const float* x      = (const float*)d_in[0];
    const int*   ei     = (const int*)d_in[1];
    const int*   pbi    = (const int*)d_in[2];
    const int*   iei    = (const int*)d_in[3];
    const float* ieattr = (const float*)d_in[4];
    const int*   cb     = (const int*)d_in[5];

    // params flattened in jax pytree (sorted-key) order, appended after inputs
    const float* F[80];
    for (int i = 6; i < n_in && i < 86; ++i) F[i - 6] = (const float*)d_in[i];
    #define ECC(l, j)   F[(l)*8 + (j)]
    #define GATW(l)     F[24 + (l)*4 + 0]
    #define GATAD(l)    F[24 + (l)*4 + 1]
    #define GATAS(l)    F[24 + (l)*4 + 2]
    #define GATB(l)     F[24 + (l)*4 + 3]
    const float *gp1W = F[36], *gp1b = F[37], *gp2W = F[38], *gp2b = F[39];
    const float *h1W = F[40], *h1b = F[41], *h2W = F[42], *h2b = F[43], *h3W = F[44], *h3b = F[45];
    const float *iinW = F[46], *iinb = F[47];
    #define ILNBE(l)  F[48 + (l)*2]
    #define ILNG(l)   F[49 + (l)*2]
    const float *ioutW = F[54], *ioutb = F[55];
    const float *attW = F[56], *attb = F[57];
    const float *xinW = F[58], *xinb = F[59];
    #define ALNBE(l)  F[60 + (l)*2]
    #define ALNG(l)   F[61 + (l)*2]
    const float *xoutW = F[66], *xoutb = F[67];

    // ---- workspace layout (floats) ----
    float* ws = (float*)d_ws;
    size_t o = 0;
    float* h      = ws + o; o += (size_t)N_ATOMS * Hd;
    float* hn     = ws + o; o += (size_t)N_ATOMS * Hd;
    float* wx     = ws + o; o += (size_t)N_ATOMS * HEADS * Hd;   // aliased in stage 2
    float* as_    = ws + o; o += (size_t)N_ATOMS * HEADS;
    float* ad_    = ws + o; o += (size_t)N_ATOMS * HEADS;
    float* segmax = ws + o; o += (size_t)N_ATOMS * HEADS;
    float* segsum = ws + o; o += (size_t)N_ATOMS * HEADS;
    float* ealpha = ws + o; o += (size_t)ET * HEADS;
    float* attw   = ws + o; o += (size_t)N_ATOMS;
    float* red    = ws + o; o += 8;
    float* poly   = ws + o; o += (size_t)N_POLY * Hd;
    float* gbuf   = ws + o; o += (size_t)N_POLY * Hd;
    float* gtmp   = ws + o; o += (size_t)N_POLY * Hd;
    float* aggr   = ws + o; o += (size_t)N_POLY * Hd;
    float* cat256 = ws + o; o += (size_t)N_POLY * 2 * Hd;
    float* u1o    = ws + o; o += (size_t)N_POLY * Hd;
    float* gout   = ws + o; o += (size_t)N_POLY * Hd;
    float* gs1    = ws + o; o += (size_t)N_POLY * (Hd / 2);
    float* gscore = ws + o; o += (size_t)N_POLY;
    float* cmax   = ws + o; o += N_CRYSTAL;
    float* csum   = ws + o; o += N_CRYSTAL;
    float* graph  = ws + o; o += (size_t)N_CRYSTAL * Hd;
    float* hh1    = ws + o; o += (size_t)N_CRYSTAL * (Hd / 2);
    float* hh2    = ws + o; o += (size_t)N_CRYSTAL * (Hd / 4);
    // stage-2 edge buffers alias wx (dead after stage 1)
    float* cat260 = wx;
    float* msg1   = wx + (size_t)E_POLY * (2 * Hd + EDGE_DIM);
    float* msg2   = msg1 + (size_t)E_POLY * Hd;

    auto cdiv = [](long long a, long long b) { return (unsigned)((a + b - 1) / b); };
    const float NEG_INF = -__builtin_inff();

    // ================= Stage 1: intra-polyhedral GAT =================
    k_wmma_gemm<<<dim3(Hd / 32, N_ATOMS / 16), 64, 0, stream>>>(x, xinW, xinb, h,
                                                                N_ATOMS, D_IN, Hd, 1);
    for (int l = 0; l < N_LAYERS; ++l) {
        k_wmma_gemm<<<dim3(HEADS * Hd / 32, N_ATOMS / 16), 64, 0, stream>>>(
            h, GATW(l), nullptr, wx, N_ATOMS, Hd, HEADS * Hd, 0);
        k_gat_scores<<<cdiv(N_ATOMS * HEADS, 256), 256, 0, stream>>>(wx, GATAS(l), GATAD(l), as_, ad_);
        k_fill<<<cdiv(N_ATOMS * HEADS, 256), 256, 0, stream>>>(segmax, NEG_INF, N_ATOMS * HEADS);
        k_fill<<<cdiv(N_ATOMS * HEADS, 256), 256, 0, stream>>>(segsum, 0.f, N_ATOMS * HEADS);
        k_fill<<<cdiv((long long)N_ATOMS * Hd, 256), 256, 0, stream>>>(hn, 0.f, N_ATOMS * Hd);
        k_edge_max<<<cdiv((long long)ET * HEADS, 256), 256, 0, stream>>>(ei, as_, ad_, segmax);
        k_edge_exp<<<cdiv((long long)ET * HEADS, 256), 256, 0, stream>>>(ei, as_, ad_, segmax, segsum, ealpha);
        k_edge_aggr<<<cdiv((long long)ET * Hd, 256), 256, 0, stream>>>(ei, ealpha, segsum, wx, hn);
        k_ln_relu_res<<<cdiv(N_ATOMS, 8), 256, 0, stream>>>(hn, GATB(l), ALNG(l), ALNBE(l), h, N_ATOMS);
    }
    // global-softmax attention pooling -> poly
    k_gemv<<<cdiv(N_ATOMS, 256), 256, 0, stream>>>(h, attW, attb, attw, N_ATOMS, Hd);
    k_fill<<<1, 32, 0, stream>>>(red, NEG_INF, 1);
    k_fill<<<1, 32, 0, stream>>>(red + 1, 0.f, 1);
    k_reduce_max<<<64, 256, 0, stream>>>(attw, red, N_ATOMS);
    k_exp_sum<<<cdiv(N_ATOMS, 256), 256, 0, stream>>>(attw, red, N_ATOMS);
    k_fill<<<cdiv(N_POLY * Hd, 256), 256, 0, stream>>>(poly, 0.f, N_POLY * Hd);
    k_attpool<<<cdiv((long long)N_ATOMS * Hd, 256), 256, 0, stream>>>(h, attw, red, pbi, poly);
    k_wmma_gemm<<<dim3(Hd / 32, N_POLY / 16), 64, 0, stream>>>(poly, xoutW, xoutb, gtmp,
                                                               N_POLY, Hd, Hd, 0);
    k_wmma_gemm<<<dim3(Hd / 32, N_POLY / 16), 64, 0, stream>>>(gtmp, iinW, iinb, gbuf,
                                                               N_POLY, Hd, Hd, 1);

    // ================= Stage 2: inter-polyhedral ECC =================
    for (int l = 0; l < N_LAYERS; ++l) {
        k_cat260<<<cdiv((long long)E_POLY * (2 * Hd + EDGE_DIM), 256), 256, 0, stream>>>(
            gbuf, iei, ieattr, cat260);
        k_wmma_gemm<<<dim3(Hd / 32, E_POLY / 16), 64, 0, stream>>>(
            cat260, ECC(l, 0), ECC(l, 1), msg1, E_POLY, 2 * Hd + EDGE_DIM, Hd, 1);
        k_wmma_gemm<<<dim3(Hd / 32, E_POLY / 16), 64, 0, stream>>>(
            msg1, ECC(l, 2), ECC(l, 3), msg2, E_POLY, Hd, Hd, 0);
        k_fill<<<cdiv(N_POLY * Hd, 256), 256, 0, stream>>>(aggr, 0.f, N_POLY * Hd);
        k_scatter_add<<<cdiv((long long)E_POLY * Hd, 256), 256, 0, stream>>>(iei, msg2, aggr);
        k_cat256<<<cdiv(N_POLY * 2 * Hd, 256), 256, 0, stream>>>(gbuf, aggr, cat256);
        k_wmma_gemm<<<dim3(Hd / 32, N_POLY / 16), 64, 0, stream>>>(
            cat256, ECC(l, 4), ECC(l, 5), u1o, N_POLY, 2 * Hd, Hd, 1);
        k_wmma_gemm<<<dim3(Hd / 32, N_POLY / 16), 64, 0, stream>>>(
            u1o, ECC(l, 6), ECC(l, 7), gtmp, N_POLY, Hd, Hd, 0);
        k_ln_relu_res<<<cdiv(N_POLY, 8), 256, 0, stream>>>(gtmp, nullptr, ILNG(l), ILNBE(l), gbuf, N_POLY);
    }
    k_wmma_gemm<<<dim3(Hd / 32, N_POLY / 16), 64, 0, stream>>>(gbuf, ioutW, ioutb, gout,
                                                               N_POLY, Hd, Hd, 0);

    // ================= crystal pooling + head =================
    k_wmma_gemm<<<dim3((Hd / 2) / 32, N_POLY / 16), 64, 0, stream>>>(gout, gp1W, gp1b, gs1,
                                                                     N_POLY, Hd, Hd / 2, 1);
    k_gemv<<<cdiv(N_POLY, 256), 256, 0, stream>>>(gs1, gp2W, gp2b, gscore, N_POLY, Hd / 2);
    k_fill<<<1, 32, 0, stream>>>(cmax, NEG_INF, N_CRYSTAL);
    k_fill<<<1, 32, 0, stream>>>(csum, 0.f, N_CRYSTAL);
    k_seg_max<<<cdiv(N_POLY, 256), 256, 0, stream>>>(gscore, cb, cmax, N_POLY);
    k_fix_max<<<1, N_CRYSTAL, 0, stream>>>(cmax, N_CRYSTAL);
    k_seg_exp<<<cdiv(N_POLY, 256), 256, 0, stream>>>(gscore, cb, cmax, csum, N_POLY);
    k_fill<<<cdiv(N_CRYSTAL * Hd, 256), 256, 0, stream>>>(graph, 0.f, N_CRYSTAL * Hd);
    k_cpool<<<cdiv(N_POLY * Hd, 256), 256, 0, stream>>>(gout, gscore, csum, cb, graph);

    k_wmma_gemm<<<dim3((Hd / 2) / 32, N_CRYSTAL / 16), 64, 0, stream>>>(
        graph, h1W, h1b, hh1, N_CRYSTAL, Hd, Hd / 2, 1);
    k_wmma_gemm<<<dim3((Hd / 4) / 32, N_CRYSTAL / 16), 64, 0, stream>>>(
        hh1, h2W, h2b, hh2, N_CRYSTAL, Hd / 2, Hd / 4, 1);
    k_gemv<<<1, 32, 0, stream>>>(hh2, h3W, h3b, (float*)d_out, N_CRYSTAL, Hd / 4);
}